// GraphTransformer_48558900249039
// MI455X (gfx1250) — compile-verified
//
#include <hip/hip_runtime.h>
#include <math.h>

#define NNODES 4096
#define DIM    256
#define NHEADS 8
#define HC     2048   // H * C
#define MAXD   192    // degree cap; Binomial(4096, 0.01) => mean 41, P(deg>192) ~ 0
#define LN_EPS 1e-5f

typedef __attribute__((ext_vector_type(2))) float v2f;
typedef __attribute__((ext_vector_type(8))) float v8f;

// ---------------------------------------------------------------------------
// Kernel 1: deterministic neighbor-list build.
// One block per target node i; scans adjacency column adj[j][i] in j order and
// compacts edge sources with an ordered block prefix scan (no atomics =>
// bitwise-deterministic neighbor order => deterministic fp sums downstream).
// ---------------------------------------------------------------------------
__global__ __launch_bounds__(256) void build_nbr_kernel(
    const float* __restrict__ adj, int* __restrict__ deg, int* __restrict__ nbr)
{
    __shared__ int scanbuf[256];
    __shared__ int cntS;
    const int i   = blockIdx.x;
    const int tid = threadIdx.x;
    if (tid == 0) cntS = 0;
    __syncthreads();

    for (int base = 0; base < NNODES; base += 256) {
        const int   j    = base + tid;
        const float a    = adj[(size_t)j * NNODES + i];   // adj[j,i]!=0 => edge j->i
        const int   flag = (a != 0.0f) ? 1 : 0;
        scanbuf[tid] = flag;
        __syncthreads();
        // Hillis-Steele inclusive scan over 256 threads
        for (int s = 1; s < 256; s <<= 1) {
            int v = (tid >= s) ? scanbuf[tid - s] : 0;
            __syncthreads();
            if (tid >= s) scanbuf[tid] += v;
            __syncthreads();
        }
        const int pos = cntS + scanbuf[tid] - flag;   // exclusive prefix + running count
        if (flag && pos < MAXD) nbr[i * MAXD + pos] = j;
        __syncthreads();
        if (tid == 255) cntS += scanbuf[255];
    }
    __syncthreads();
    if (tid == 0) deg[i] = (cntS < MAXD) ? cntS : MAXD;
}

// ---------------------------------------------------------------------------
// Kernel 2: fp32 WMMA GEMM  Out[4096 x Nout] = X[4096 x 256] @ W[256 x Nout] + b
// Block = 256 threads = 8 waves; block tile = 64(M) x 128(N).
// Each wave owns a 16-column slice and FOUR 16x16 M-subtiles, so every
// B-fragment (2 global_load_b32) is amortized over 4 V_WMMA_F32_16X16X4_F32
// ops (WMMA:VMEM issue ratio 2:1 instead of 1:2), and the 64 KB LDS A-tile is
// reused 4x per wave.
// fp32 A-frag layout: lanes 0-15 K={0,1}, lanes 16-31 K={2,3} (VGPR0,VGPR1).
// B 4x16 uses the same K split; C/D: VGPR r -> row r + 8*half, col = lane&15.
// ---------------------------------------------------------------------------
__global__ __launch_bounds__(256) void gemm_wmma_f32_kernel(
    const float* __restrict__ X, const float* __restrict__ W,
    const float* __restrict__ bias, float* __restrict__ Out, int Nout)
{
    __shared__ float xs[64 * DIM];   // 64 KB A tile, shared by the 8 waves
    const int tid   = threadIdx.x;
    const int mBase = blockIdx.y * 64;

    // vectorized A-tile staging: global_load_b128 -> ds_store_b128
    {
        const float4* Xv  = (const float4*)(X + (size_t)mBase * DIM);
        float4*       xsv = (float4*)xs;
        #pragma unroll
        for (int idx = tid; idx < (64 * DIM) / 4; idx += 256)
            xsv[idx] = Xv[idx];
    }
    __syncthreads();

    const int wave = tid >> 5;
    const int lane = tid & 31;
    const int half = lane >> 4;                  // 0: lanes 0-15, 1: lanes 16-31
    const int m    = lane & 15;
    const int col  = blockIdx.x * 128 + wave * 16 + m;

    const float* wb = W + col;                   // column pointer into W[256][Nout]
    v8f c0 = {0.f,0.f,0.f,0.f,0.f,0.f,0.f,0.f};
    v8f c1 = c0, c2 = c0, c3 = c0;

    #pragma unroll 4
    for (int kk = 0; kk < 64; ++kk) {
        const int k0 = kk * 4 + half * 2;
        v2f b;
        b.x = wb[(size_t)(k0)     * Nout];
        b.y = wb[(size_t)(k0 + 1) * Nout];

        v2f a0, a1, a2, a3;
        a0.x = xs[(m     ) * DIM + k0]; a0.y = xs[(m     ) * DIM + k0 + 1];
        a1.x = xs[(m + 16) * DIM + k0]; a1.y = xs[(m + 16) * DIM + k0 + 1];
        a2.x = xs[(m + 32) * DIM + k0]; a2.y = xs[(m + 32) * DIM + k0 + 1];
        a3.x = xs[(m + 48) * DIM + k0]; a3.y = xs[(m + 48) * DIM + k0 + 1];

        // (neg_a, A, neg_b, B, c_mod, C, reuse_a, reuse_b)
        c0 = __builtin_amdgcn_wmma_f32_16x16x4_f32(false, a0, false, b, (short)0, c0, false, false);
        c1 = __builtin_amdgcn_wmma_f32_16x16x4_f32(false, a1, false, b, (short)0, c1, false, false);
        c2 = __builtin_amdgcn_wmma_f32_16x16x4_f32(false, a2, false, b, (short)0, c2, false, false);
        c3 = __builtin_amdgcn_wmma_f32_16x16x4_f32(false, a3, false, b, (short)0, c3, false, false);
    }

    const float bcol = bias[col];
    #pragma unroll
    for (int r = 0; r < 8; ++r) {
        const int row = mBase + r + 8 * half;
        Out[(size_t)(row     ) * Nout + col] = c0[r] + bcol;
        Out[(size_t)(row + 16) * Nout + col] = c1[r] + bcol;
        Out[(size_t)(row + 32) * Nout + col] = c2[r] + bcol;
        Out[(size_t)(row + 48) * Nout + col] = c3[r] + bcol;
    }
}

// ---------------------------------------------------------------------------
// Kernel 3: fused sparse attention + head-mean + skip + LayerNorm + residual.
// One block per target node; 8 waves = 8 heads (wave32). K/V gathers are
// L2-resident (each is 32 MB << 192 MB global L2).
// ---------------------------------------------------------------------------
__global__ __launch_bounds__(256) void attn_fused_kernel(
    const float* __restrict__ x,    const float* __restrict__ Q,
    const float* __restrict__ K,    const float* __restrict__ V,
    const float* __restrict__ skip, const int* __restrict__ deg,
    const int* __restrict__ nbr,    const float* __restrict__ gamma,
    const float* __restrict__ beta, float* __restrict__ out)
{
    __shared__ float sc[NHEADS * MAXD];     // per-head edge scores / weights
    __shared__ float aggLDS[NHEADS * DIM];  // per-head aggregated messages
    __shared__ float red[256];              // LayerNorm reduction buffer

    const int i    = blockIdx.x;
    const int tid  = threadIdx.x;
    const int h    = tid >> 5;              // wave id == head id
    const int lane = tid & 31;
    const int d    = min(deg[i], MAXD);
    float*      msc  = sc + h * MAXD;
    const int*  nrow = nbr + i * MAXD;

    // q_i for this head: 256 channels, 8 per lane
    float qreg[8];
    #pragma unroll
    for (int t = 0; t < 8; ++t)
        qreg[t] = Q[(size_t)i * HC + h * DIM + lane + 32 * t];

    // masked scores: s_n = <q_i, k_j> / sqrt(256)
    for (int n = 0; n < d; ++n) {
        const int    j  = nrow[n];
        const float* kj = K + (size_t)j * HC + h * DIM + lane;
        float p = 0.f;
        #pragma unroll
        for (int t = 0; t < 8; ++t) p += qreg[t] * kj[32 * t];
        #pragma unroll
        for (int off = 16; off > 0; off >>= 1) p += __shfl_down(p, off, 32);
        if (lane == 0) msc[n] = p * 0.0625f;   // 1/sqrt(256)
    }
    __builtin_amdgcn_wave_barrier();

    // wave-local softmax over the <=MAXD edge scores
    float mx = -3.4e38f;
    for (int n = lane; n < d; n += 32) mx = fmaxf(mx, msc[n]);
    #pragma unroll
    for (int off = 16; off > 0; off >>= 1) mx = fmaxf(mx, __shfl_xor(mx, off, 32));
    float sum = 0.f;
    for (int n = lane; n < d; n += 32) {
        const float e = __expf(msc[n] - mx);
        msc[n] = e;
        sum += e;
    }
    #pragma unroll
    for (int off = 16; off > 0; off >>= 1) sum += __shfl_xor(sum, off, 32);
    const float inv = (d > 0) ? 1.f / sum : 0.f;  // no in-edges => zero messages
    __builtin_amdgcn_wave_barrier();

    // weighted aggregation of v_j
    float acc[8] = {0.f, 0.f, 0.f, 0.f, 0.f, 0.f, 0.f, 0.f};
    for (int n = 0; n < d; ++n) {
        const int    j  = nrow[n];
        const float  p  = msc[n] * inv;
        const float* vj = V + (size_t)j * HC + h * DIM + lane;
        #pragma unroll
        for (int t = 0; t < 8; ++t) acc[t] += p * vj[32 * t];
    }
    #pragma unroll
    for (int t = 0; t < 8; ++t) aggLDS[h * DIM + lane + 32 * t] = acc[t];
    __syncthreads();

    // head mean + skip GEMM result
    const int c = tid;
    float val = 0.f;
    #pragma unroll
    for (int hh = 0; hh < NHEADS; ++hh) val += aggLDS[hh * DIM + c];
    val = val * 0.125f + skip[(size_t)i * DIM + c];

    // LayerNorm over 256 channels
    red[c] = val;
    __syncthreads();
    for (int s = 128; s > 0; s >>= 1) { if (c < s) red[c] += red[c + s]; __syncthreads(); }
    const float mu = red[0] * (1.f / 256.f);
    __syncthreads();
    const float dv = val - mu;
    red[c] = dv * dv;
    __syncthreads();
    for (int s = 128; s > 0; s >>= 1) { if (c < s) red[c] += red[c + s]; __syncthreads(); }
    const float var = red[0] * (1.f / 256.f);

    const float nrm = dv * rsqrtf(var + LN_EPS) * gamma[c] + beta[c];
    out[(size_t)i * DIM + c] = nrm + x[(size_t)i * DIM + c];   // residual
}

// ---------------------------------------------------------------------------
extern "C" void kernel_launch(void* const* d_in, const int* in_sizes, int n_in,
                              void* d_out, int out_size, void* d_ws, size_t ws_size,
                              hipStream_t stream)
{
    (void)in_sizes; (void)n_in; (void)out_size; (void)ws_size;
    const float* x     = (const float*)d_in[0];
    const float* adj   = (const float*)d_in[1];
    const float* Wq    = (const float*)d_in[2];
    const float* bq    = (const float*)d_in[3];
    const float* Wk    = (const float*)d_in[4];
    const float* bk    = (const float*)d_in[5];
    const float* Wv    = (const float*)d_in[6];
    const float* bv    = (const float*)d_in[7];
    const float* Wsk   = (const float*)d_in[8];
    const float* bsk   = (const float*)d_in[9];
    const float* gamma = (const float*)d_in[10];
    const float* beta  = (const float*)d_in[11];
    float* out = (float*)d_out;

    // workspace layout (bytes)
    char*  ws  = (char*)d_ws;
    float* Q   = (float*)(ws);                           // 32 MB
    float* Kb  = (float*)(ws + 33554432ull);             // 32 MB
    float* Vb  = (float*)(ws + 67108864ull);             // 32 MB
    float* SK  = (float*)(ws + 100663296ull);            //  4 MB
    int*   deg = (int*)  (ws + 104857600ull);            // 16 KB
    int*   nbr = (int*)  (ws + 104857600ull + 16384ull); //  3 MB

    build_nbr_kernel<<<NNODES, 256, 0, stream>>>(adj, deg, nbr);

    gemm_wmma_f32_kernel<<<dim3(HC / 128, NNODES / 64), 256, 0, stream>>>(x, Wq,  bq,  Q,  HC);
    gemm_wmma_f32_kernel<<<dim3(HC / 128, NNODES / 64), 256, 0, stream>>>(x, Wk,  bk,  Kb, HC);
    gemm_wmma_f32_kernel<<<dim3(HC / 128, NNODES / 64), 256, 0, stream>>>(x, Wv,  bv,  Vb, HC);
    gemm_wmma_f32_kernel<<<dim3(DIM / 128, NNODES / 64), 256, 0, stream>>>(x, Wsk, bsk, SK, DIM);

    attn_fused_kernel<<<NNODES, 256, 0, stream>>>(x, Q, Kb, Vb, SK, deg, nbr, gamma, beta, out);
}